// GroupedConv_12704513261872
// MI455X (gfx1250) — compile-verified
//
#include <hip/hip_runtime.h>

// CDNA5 / gfx1250 grouped 3x3 conv via V_WMMA_F32_16X16X4_F32.
// Per-wave tile: D(16x16) = A(16x144) x B(144x16) + C.
// K re-ordered as k = tap*16 + sel*8 + ci (shifts/ands only, no divisions).
// The block-diagonal A matrix is staged in LDS WITH its zeros baked in, so the
// hot loop needs no weight masking: 1 ds_load_b64 + 2 global_load_b32 +
// 2 boundary cndmasks per WMMA, fully unrolled 9 taps x 4 K-steps.

typedef float v2f __attribute__((ext_vector_type(2)));
typedef float v8f __attribute__((ext_vector_type(8)));

#define BATCH  32
#define CIN    256
#define IMG_H  56
#define IMG_W  56
#define HWSZ   (IMG_H * IMG_W)      // 3136 = 196 * 16
#define CIN_G  8                    // input channels per group
#define OC_G   8                    // output channels per group
#define KTAPS  9                    // 3x3
#define KG     (CIN_G * KTAPS)      // 72 weights per output channel
#define KPAIR  144                  // K per fused group-pair (2 groups x 72)
#define PAIRS_PER_BLOCK 4           // 4 waves/block, one group-pair per wave
#define THREADS 128
#define A_LDS_FLOATS (PAIRS_PER_BLOCK * 16 * KPAIR)   // 9216 floats = 36 KB

__global__ __launch_bounds__(THREADS)
void grouped_conv3x3_wmma(const float* __restrict__ x,
                          const float* __restrict__ w,
                          float* __restrict__ out)
{
    // Full block-diagonal A: [pair(4)][m(16)][k(144)], zeros included.
    __shared__ float a_lds[A_LDS_FLOATS];

    const int tid     = threadIdx.x;
    const int tile    = blockIdx.x;   // 0..195 : 16-pixel tile within one image
    const int pairBlk = blockIdx.y;   // 0..3   : which 4 group-pairs
    const int img     = blockIdx.z;   // 0..31

    // ---- Build block-diagonal A in LDS (one-time, w is L2-resident).
    // 64 rows of 144; 128 threads -> each thread fills one (row, K-half).
    {
        const int row   = tid >> 1;          // 0..63 : pr*16 + m
        const int halfk = tid & 1;           // which 72-wide K half
        const int pr    = row >> 4;
        const int m     = row & 15;
        const int mHi   = (m >= OC_G) ? 1 : 0;
        // global oc for this row = (pairBlk*4 + pr)*16 + m
        const float* __restrict__ wr =
            w + (size_t)(((pairBlk * PAIRS_PER_BLOCK + pr) << 4) + m) * KG;
        float* __restrict__ dst = &a_lds[row * KPAIR + halfk * (KPAIR / 2)];
        for (int kk = 0; kk < KPAIR / 2; ++kk) {
            const int k   = halfk * (KPAIR / 2) + kk;
            const int tap = k >> 4;          // k = tap*16 + sel*8 + ci
            const int sel = (k >> 3) & 1;
            const int ci  = k & 7;
            const float v = wr[ci * KTAPS + tap];
            dst[kk] = (mHi == sel) ? v : 0.0f;
        }
    }
    __syncthreads();

    const int wave = tid >> 5;        // group-pair slot within block
    const int lane = tid & 31;
    const int hlf  = lane >> 4;       // 0: lanes 0-15, 1: lanes 16-31
    const int mn   = lane & 15;       // A-row (m) / B-col (n) for this lane

    const int gp = pairBlk * PAIRS_PER_BLOCK + wave;   // global group-pair 0..15

    // Per-lane A base: [pair][m][.] plus the half-wave K shift (+2*hlf).
    const float* __restrict__ aBase =
        &a_lds[(wave * 16 + mn) * KPAIR + 2 * hlf];

    // B-side pixel for this lane (N dim): 16 contiguous linear HW indices.
    const int p  = tile * 16 + mn;
    const int ph = p / IMG_W;
    const int pw = p - ph * IMG_W;

    // Per-lane B channel base: includes group-pair and the half-wave ci shift.
    const float* __restrict__ xBase =
        x + ((size_t)img * CIN + (size_t)(gp * 16 + 2 * hlf)) * HWSZ;

    v8f acc = {};   // C/D: 16x16 f32, 8 VGPRs

    #pragma unroll
    for (int t = 0; t < KTAPS; ++t) {
        const int dh = t / 3 - 1;             // compile-time after unroll
        const int dw = t - (t / 3) * 3 - 1;

        // One clamped address + one validity mask per tap, shared by 8 loads.
        const int hh = ph + dh;
        const int ww = pw + dw;
        const bool ok = ((unsigned)hh < IMG_H) & ((unsigned)ww < IMG_W);
        const int hc = (hh < 0) ? 0 : ((hh > IMG_H - 1) ? IMG_H - 1 : hh);
        const int wc = (ww < 0) ? 0 : ((ww > IMG_W - 1) ? IMG_W - 1 : ww);
        const float* __restrict__ bp = xBase + hc * IMG_W + wc;

        #pragma unroll
        for (int s = 0; s < 4; ++s) {
            const int sel    = s >> 1;                 // compile-time
            const int ciBase = (s & 1) * 4;            // compile-time
            const int chan   = sel * OC_G + ciBase;    // rel. channel offset

            // ---- A fragment: one ds_load_b64, zeros already baked in.
            const v2f a = *(const v2f*)(aBase + t * 16 + s * 4);

            // ---- B fragment: two b32 loads at immediate channel offsets.
            const float b0 = bp[(size_t)(chan + 0) * HWSZ];
            const float b1 = bp[(size_t)(chan + 1) * HWSZ];
            v2f b;
            b[0] = ok ? b0 : 0.0f;
            b[1] = ok ? b1 : 0.0f;

            // (neg_a, A, neg_b, B, c_mod, C, reuse_a, reuse_b)
            acc = __builtin_amdgcn_wmma_f32_16x16x4_f32(
                      false, a, false, b, (short)0, acc, false, false);
        }
    }

    // ---- Store D: lane (hlf,mn) holds rows m = v + 8*hlf, col n = mn.
    // out is NCHW: out[img][gp*16 + m][p]; 16-lane-contiguous along p.
    float* __restrict__ ob =
        out + ((size_t)img * CIN + (size_t)gp * 16) * HWSZ + p;
    #pragma unroll
    for (int v = 0; v < 8; ++v) {
        const int m = v + 8 * hlf;
        ob[(size_t)m * HWSZ] = acc[v];
    }
}

extern "C" void kernel_launch(void* const* d_in, const int* in_sizes, int n_in,
                              void* d_out, int out_size, void* d_ws, size_t ws_size,
                              hipStream_t stream) {
    const float* x = (const float*)d_in[0];   // [32,256,56,56] f32
    const float* w = (const float*)d_in[1];   // [256,8,3,3]    f32
    float* out = (float*)d_out;               // [32,256,56,56] f32
    (void)in_sizes; (void)n_in; (void)out_size; (void)d_ws; (void)ws_size;

    dim3 grid(HWSZ / 16, /*pair quarters=*/4, BATCH);   // (196, 4, 32)
    dim3 block(THREADS);                                 // 4 wave32 waves
    grouped_conv3x3_wmma<<<grid, block, 0, stream>>>(x, w, out);
}